// GNNBackbone_8143257993843
// MI455X (gfx1250) — compile-verified
//
#include <hip/hip_runtime.h>
#include <hip/hip_bf16.h>

// ---------------------------------------------------------------------------
// GCN backbone for MI455X (gfx1250, wave32).
//  h0 = x @ w_in + b_in
//  t  = h0 @ w_conv1 ; acc = segment_sum(t[src]*norm, dst) (incl self loops)
//  h0 = relu(acc + b_conv1)
//  t  = h0 @ w_conv2 ; acc = segment_sum(...)
//  out = relu(acc + b_conv2)
// GEMMs use V_WMMA_F32_16X16X4_F32 (full fp32 accuracy; compute is nowhere
// near the bottleneck — edge scatter through L2 is).
// ---------------------------------------------------------------------------

typedef __attribute__((ext_vector_type(2))) float v2f;
typedef __attribute__((ext_vector_type(8))) float v8f;

#define HIDF 128
#define INNF 64

// ---- degree / normalization -------------------------------------------------
__global__ __launch_bounds__(256)
void deg_init_kernel(float* __restrict__ deg, int n) {
    int i = blockIdx.x * blockDim.x + threadIdx.x;
    if (i < n) deg[i] = 1.0f;   // self loop contributes 1 to every node
}

__global__ __launch_bounds__(256)
void deg_accum_kernel(const int* __restrict__ dst, float* __restrict__ deg, int e) {
    int i = blockIdx.x * blockDim.x + threadIdx.x;
    if (i < e) atomicAdd(&deg[dst[i]], 1.0f);
}

__global__ __launch_bounds__(256)
void deg_rsqrt_kernel(float* __restrict__ deg, int n) {
    int i = blockIdx.x * blockDim.x + threadIdx.x;
    if (i < n) deg[i] = rsqrtf(deg[i]);   // deg >= 1 always (self loop)
}

// ---- fp32 WMMA GEMM: C[M,N] = A[M,K] @ W[K,N] (+ bias) ----------------------
// One wave per 16x16 output tile. ISA VGPR layouts (CDNA5 ISA):
//   A 16x4 : lane = half*16 + r, VGPR v = A[r][k0 + v + 2*half]
//   B 4x16 : lane = half*16 + r, VGPR v = W[k0 + v + 2*half][col r]
//   D 16x16: lane = half*16 + r, VGPR v = C[v + 8*half][col r]
__global__ __launch_bounds__(256)
void gemm_wmma_f32_kernel(const float* __restrict__ A, const float* __restrict__ W,
                          const float* __restrict__ bias, float* __restrict__ C,
                          int M, int K, int N)
{
    const int wave = threadIdx.x >> 5;
    const int lane = threadIdx.x & 31;
    const int half = lane >> 4;
    const int lr   = lane & 15;

    const int Mt = (M + 15) >> 4;
    const int Nt = N >> 4;
    const int tile = blockIdx.x * 8 + wave;
    if (tile >= Mt * Nt) return;
    const int tm = tile / Nt;
    const int tn = tile - tm * Nt;

    int arow = tm * 16 + lr;
    if (arow > M - 1) arow = M - 1;            // clamp (keeps EXEC all-ones)
    const int bcol = tn * 16 + lr;

    const float* __restrict__ Arow = A + (size_t)arow * K;
    const float* __restrict__ Wcol = W + bcol;

    v8f acc = {0.f, 0.f, 0.f, 0.f, 0.f, 0.f, 0.f, 0.f};

    for (int k = 0; k < K; k += 4) {
        const int ka = k + 2 * half;
        v2f a = *(const v2f*)(Arow + ka);       // A[r][ka], A[r][ka+1] (8B aligned)
        v2f b;
        b.x = Wcol[(size_t)(ka)     * N];
        b.y = Wcol[(size_t)(ka + 1) * N];
        acc = __builtin_amdgcn_wmma_f32_16x16x4_f32(
            /*neg_a=*/false, a, /*neg_b=*/false, b,
            /*c_mod=*/(short)0, acc, /*reuse_a=*/false, /*reuse_b=*/false);
    }

    const float bv = bias ? bias[bcol] : 0.0f;
#pragma unroll
    for (int v = 0; v < 8; ++v) {
        const int m = tm * 16 + v + 8 * half;
        if (m < M) C[(size_t)m * N + bcol] = acc[v] + bv;
    }
}

// ---- self-loop init: acc[i][:] = t[i][:] * dinv[i]^2 ------------------------
__global__ __launch_bounds__(256)
void acc_selfloop_kernel(const float* __restrict__ t, const float* __restrict__ dinv,
                         float* __restrict__ acc, int n)
{
    int i4 = blockIdx.x * blockDim.x + threadIdx.x;   // float4 index
    int tot = n * (HIDF / 4);
    if (i4 >= tot) return;
    int node = i4 / (HIDF / 4);
    float d = dinv[node];
    float s = d * d;
    float4 v = ((const float4*)t)[i4];
    v.x *= s; v.y *= s; v.z *= s; v.w *= s;
    ((float4*)acc)[i4] = v;
}

// ---- edge scatter: acc[dst] += t[src] * dinv[src]*dinv[dst] -----------------
// One wave (32 lanes) per edge; each lane moves 4 consecutive floats.
__global__ __launch_bounds__(256)
void edge_scatter_kernel(const int* __restrict__ src, const int* __restrict__ dst,
                         const float* __restrict__ dinv, const float* __restrict__ t,
                         float* __restrict__ acc, int e)
{
    const int eid = blockIdx.x * 8 + (threadIdx.x >> 5);
    if (eid >= e) return;
    const int lane = threadIdx.x & 31;
    const int s = src[eid];
    const int d = dst[eid];
    const float nrm = dinv[s] * dinv[d];

    float4 v = ((const float4*)(t + (size_t)s * HIDF))[lane];
    float* ap = acc + (size_t)d * HIDF + lane * 4;
    atomicAdd(ap + 0, v.x * nrm);
    atomicAdd(ap + 1, v.y * nrm);
    atomicAdd(ap + 2, v.z * nrm);
    atomicAdd(ap + 3, v.w * nrm);
}

// ---- finalize: out = relu(acc + bias) ---------------------------------------
__global__ __launch_bounds__(256)
void finalize_relu_kernel(const float* __restrict__ acc, const float* __restrict__ bias,
                          float* __restrict__ out, int n)
{
    int i4 = blockIdx.x * blockDim.x + threadIdx.x;
    int tot = n * (HIDF / 4);
    if (i4 >= tot) return;
    int j0 = (i4 & (HIDF / 4 - 1)) * 4;
    float4 b = *(const float4*)(bias + j0);
    float4 v = ((const float4*)acc)[i4];
    v.x = fmaxf(v.x + b.x, 0.0f);
    v.y = fmaxf(v.y + b.y, 0.0f);
    v.z = fmaxf(v.z + b.z, 0.0f);
    v.w = fmaxf(v.w + b.w, 0.0f);
    ((float4*)out)[i4] = v;
}

// ---------------------------------------------------------------------------
extern "C" void kernel_launch(void* const* d_in, const int* in_sizes, int n_in,
                              void* d_out, int out_size, void* d_ws, size_t ws_size,
                              hipStream_t stream)
{
    const float* x    = (const float*)d_in[0];
    const int*   ei   = (const int*)  d_in[1];
    const float* w_in = (const float*)d_in[2];
    const float* b_in = (const float*)d_in[3];
    const float* w1   = (const float*)d_in[4];
    const float* b1   = (const float*)d_in[5];
    const float* w2   = (const float*)d_in[6];
    const float* b2   = (const float*)d_in[7];

    const int N = in_sizes[0] / INNF;     // 50000
    const int E = in_sizes[1] / 2;        // 600000
    const int* src = ei;
    const int* dst = ei + E;

    // workspace layout
    char* ws = (char*)d_ws;
    float* dinv = (float*)ws;                                   // N floats (deg then rsqrt)
    size_t off  = (((size_t)N * 4) + 255) & ~(size_t)255;
    size_t hsz  = (size_t)N * HIDF * sizeof(float);
    float* h0   = (float*)(ws + off);                           // N*HID
    float* t1   = (float*)(ws + off + hsz);                     // N*HID
    float* accb = (float*)(ws + off + 2 * hsz);                 // N*HID
    float* out  = (float*)d_out;

    const int THREADS = 256;
    const int nBlkN   = (N + THREADS - 1) / THREADS;
    const int nBlkE   = (E + THREADS - 1) / THREADS;
    const int nBlkV4  = (N * (HIDF / 4) + THREADS - 1) / THREADS;
    const int nBlkEw  = (E + 7) / 8;                            // wave per edge
    const int tilesIn = ((N + 15) / 16) * (HIDF / 16);
    const int nBlkGemm = (tilesIn + 7) / 8;                     // 8 waves / block

    // 1) degrees -> deg^{-1/2}
    deg_init_kernel<<<nBlkN, THREADS, 0, stream>>>(dinv, N);
    deg_accum_kernel<<<nBlkE, THREADS, 0, stream>>>(dst, dinv, E);
    deg_rsqrt_kernel<<<nBlkN, THREADS, 0, stream>>>(dinv, N);

    // 2) h0 = x @ w_in + b_in
    gemm_wmma_f32_kernel<<<nBlkGemm, THREADS, 0, stream>>>(x, w_in, b_in, h0, N, INNF, HIDF);

    // 3) conv1: t1 = h0 @ w1 ; scatter ; h0 = relu(acc + b1)
    gemm_wmma_f32_kernel<<<nBlkGemm, THREADS, 0, stream>>>(h0, w1, nullptr, t1, N, HIDF, HIDF);
    acc_selfloop_kernel<<<nBlkV4, THREADS, 0, stream>>>(t1, dinv, accb, N);
    edge_scatter_kernel<<<nBlkEw, THREADS, 0, stream>>>(src, dst, dinv, t1, accb, E);
    finalize_relu_kernel<<<nBlkV4, THREADS, 0, stream>>>(accb, b1, h0, N);

    // 4) conv2: t1 = h0 @ w2 ; scatter ; out = relu(acc + b2)
    gemm_wmma_f32_kernel<<<nBlkGemm, THREADS, 0, stream>>>(h0, w2, nullptr, t1, N, HIDF, HIDF);
    acc_selfloop_kernel<<<nBlkV4, THREADS, 0, stream>>>(t1, dinv, accb, N);
    edge_scatter_kernel<<<nBlkEw, THREADS, 0, stream>>>(src, dst, dinv, t1, accb, E);
    finalize_relu_kernel<<<nBlkV4, THREADS, 0, stream>>>(accb, b2, out, N);
}